// StructuredStateSpace_83434034692711
// MI455X (gfx1250) — compile-verified
//
#include <hip/hip_runtime.h>
#include <hip/hip_bf16.h>

// Problem constants from the reference
#define BATCH 8
#define SEQ   4096
#define DIM   1024
#define NSTATE 256
#define LN_EPS 1e-5f

typedef float v2f __attribute__((ext_vector_type(2)));
typedef float v4f __attribute__((ext_vector_type(4)));
typedef float v8f __attribute__((ext_vector_type(8)));

// ---------------------------------------------------------------------------
// Kernel 1: row means of x [rows = B*S, cols = D] via V_WMMA_F32_16X16X4_F32.
// Each wave handles 16 rows. A tile = 16x4 f32 chunk of x; B = all 1/D, so
// every column of the 16x16 accumulator D holds the row sum / D.
// A-matrix VGPR layout (32-bit 16x4): lane L<16 -> M=L, K=0,1; lane L>=16 ->
// M=L-16, K=2,3  => per lane: float2 load at (row = lane&15, col = k + 2*(L>>4)).
// ---------------------------------------------------------------------------
__global__ void __launch_bounds__(256)
ssm_rowmean_wmma(const float* __restrict__ x, float* __restrict__ xmean,
                 int rows, int cols) {
    const int wid  = (blockIdx.x * blockDim.x + threadIdx.x) >> 5;
    const int lane = threadIdx.x & 31;
    const int r0   = wid << 4;                 // 16 rows per wave
    if (r0 >= rows) return;                    // wave-uniform (never taken: exact grid)

    const int myrow = r0 + (lane & 15);
    const int koff  = (lane >> 4) << 1;        // 0 or 2
    const float* rowp = x + (size_t)myrow * cols + koff;

    const float inv = 1.0f / (float)cols;
    v2f bmat; bmat[0] = inv; bmat[1] = inv;    // constant B: layout-independent
    v8f acc = {};

#pragma unroll 8
    for (int k = 0; k < cols; k += 4) {
        v2f a = *(const v2f*)(rowp + k);       // 8B-aligned
        acc = __builtin_amdgcn_wmma_f32_16x16x4_f32(
            /*neg_a=*/false, a, /*neg_b=*/false, bmat,
            /*c_mod=*/(short)0, acc, /*reuse_a=*/false, /*reuse_b=*/false);
    }

    // C/D layout: VGPR j: lanes 0-15 -> M=j (N=lane), lanes 16-31 -> M=8+j.
    // Lane 0 holds means of rows r0..r0+7, lane 16 holds rows r0+8..r0+15.
    if (lane == 0) {
#pragma unroll
        for (int j = 0; j < 8; ++j) xmean[r0 + j] = acc[j];
    } else if (lane == 16) {
#pragma unroll
        for (int j = 0; j < 8; ++j) xmean[r0 + 8 + j] = acc[j];
    }
}

// ---------------------------------------------------------------------------
// Kernel 2: sequential scan. One wave (32 lanes) per batch; 8 states per lane
// (n = lane + 32*j). LayerNorm/dot reductions are intra-wave shuffles only.
// ---------------------------------------------------------------------------
__device__ __forceinline__ float wave_sum(float v) {
#pragma unroll
    for (int off = 16; off >= 1; off >>= 1) v += __shfl_xor(v, off, 32);
    return v;
}

__global__ void __launch_bounds__(BATCH * 32)
ssm_scan(const float* __restrict__ xmean,       // [B*S]
         const float* __restrict__ vol,         // [B*S]
         const float* __restrict__ llr,         // [N]
         const float* __restrict__ logb,        // [N]
         const float* __restrict__ cvec,        // [N]
         const float* __restrict__ log_d,       // unused here
         const float* __restrict__ log_step,    // [1]
         const float* __restrict__ vgate,       // [N]
         const float* __restrict__ lnw,         // [N]
         const float* __restrict__ lnb,         // [N]
         float* __restrict__ ys)                // [B*S]
{
    const int lane  = threadIdx.x & 31;
    const int batch = threadIdx.x >> 5;

    const float step = __expf(log_step[0]);

    float a_[8], b_[8], g_[8], w_[8], o_[8], c_[8], h_[8];
#pragma unroll
    for (int j = 0; j < 8; ++j) {
        const int n = lane + 32 * j;
        const float lam = -__expf(llr[n]);
        const float ad  = (2.0f + step * lam) * __builtin_amdgcn_rcpf(2.0f - step * lam);
        a_[j] = ad;
        b_[j] = step * (1.0f + ad) * __expf(logb[n]) * 0.5f;
        g_[j] = __builtin_amdgcn_rcpf(1.0f + __expf(-vgate[n]));   // sigmoid
        w_[j] = lnw[n];
        o_[j] = lnb[n];
        c_[j] = cvec[n];
        h_[j] = 0.0f;
    }

    const float* xmb  = xmean + batch * SEQ;
    const float* volb = vol   + batch * SEQ;
    float*       ysb  = ys    + batch * SEQ;
    const float invN = 1.0f / (float)NSTATE;

    for (int t = 0; t < SEQ; ++t) {
        const float xm = xmb[t];
        const float vt = volb[t];

        float s1 = 0.0f, s2 = 0.0f;
#pragma unroll
        for (int j = 0; j < 8; ++j) {
            const float h = fmaf(a_[j], h_[j], b_[j] * xm);
            h_[j] = h;
            s1 += h;
            s2 = fmaf(h, h, s2);
        }
        s1 = wave_sum(s1);
        s2 = wave_sum(s2);

        const float mu  = s1 * invN;
        const float var = fmaf(s2, invN, -mu * mu);
        const float rs  = __builtin_amdgcn_rsqf(var + LN_EPS);

        float dot = 0.0f;
#pragma unroll
        for (int j = 0; j < 8; ++j) {
            float hn = fmaf((h_[j] - mu) * rs, w_[j], o_[j]);
            hn *= __builtin_amdgcn_rcpf(fmaf(g_[j], vt, 1.0f));
            h_[j] = hn;
            dot = fmaf(hn, c_[j], dot);
        }
        dot = wave_sum(dot);
        if (lane == 0) ysb[t] = dot;
    }
}

// ---------------------------------------------------------------------------
// Kernel 3: out = s0 * x + s1 * ys[row], vectorized with native ext-vectors.
// s0 = a + (1-a)*e^{log_d}, s1 = (1-a), a = sigmoid(alpha).
// Non-temporal stores keep x resident in the 192MB L2 (it was loaded by K1).
// ---------------------------------------------------------------------------
__global__ void __launch_bounds__(256)
ssm_output(const float* __restrict__ x, const float* __restrict__ ys,
           const float* __restrict__ alpha, const float* __restrict__ log_d,
           float* __restrict__ out, int total4) {
    const int i = blockIdx.x * blockDim.x + threadIdx.x;
    if (i >= total4) return;

    const float a  = 1.0f / (1.0f + __expf(-alpha[0]));
    const float dd = __expf(log_d[0]);
    const float s0 = a + (1.0f - a) * dd;
    const float s1 = 1.0f - a;

    const int row = i >> 8;                 // i*4 / 1024
    const float yv = s1 * ys[row];

    const v4f xv = ((const v4f*)x)[i];
    v4f o;
    o.x = fmaf(s0, xv.x, yv);
    o.y = fmaf(s0, xv.y, yv);
    o.z = fmaf(s0, xv.z, yv);
    o.w = fmaf(s0, xv.w, yv);
    __builtin_nontemporal_store(o, ((v4f*)out) + i);
}

// ---------------------------------------------------------------------------
extern "C" void kernel_launch(void* const* d_in, const int* in_sizes, int n_in,
                              void* d_out, int out_size, void* d_ws, size_t ws_size,
                              hipStream_t stream) {
    const float* x        = (const float*)d_in[0];   // [B,S,D]
    const float* vol      = (const float*)d_in[1];   // [B,S,1]
    const float* llr      = (const float*)d_in[2];   // [N]
    const float* logb     = (const float*)d_in[3];   // [N,1]
    const float* cvec     = (const float*)d_in[4];   // [1,N]
    const float* log_d    = (const float*)d_in[5];   // [1]
    const float* log_step = (const float*)d_in[6];   // scalar
    const float* vgate    = (const float*)d_in[7];   // [N]
    const float* alpha    = (const float*)d_in[8];   // scalar
    const float* lnw      = (const float*)d_in[9];   // [N]
    const float* lnb      = (const float*)d_in[10];  // [N]
    float* out = (float*)d_out;

    const int rows = BATCH * SEQ;                    // 32768
    float* xmean = (float*)d_ws;                     // rows floats
    float* ys    = xmean + rows;                     // rows floats

    // K1: row means via WMMA. 16 rows/wave, 8 waves/block -> 128 rows/block.
    {
        const int blocks = rows / 128;               // 256, exact cover (EXEC all-1s)
        ssm_rowmean_wmma<<<blocks, 256, 0, stream>>>(x, xmean, rows, DIM);
    }
    // K2: sequential scan, one wave per batch.
    ssm_scan<<<1, BATCH * 32, 0, stream>>>(xmean, vol, llr, logb, cvec, log_d,
                                           log_step, vgate, lnw, lnb, ys);
    // K3: elementwise output, float4 vectorized.
    {
        const int total4 = (BATCH * SEQ * DIM) / 4;  // 8,388,608
        const int blocks = (total4 + 255) / 256;
        ssm_output<<<blocks, 256, 0, stream>>>(x, ys, alpha, log_d, out, total4);
    }
}